// Generator_34127810134219
// MI455X (gfx1250) — compile-verified
//
#include <hip/hip_runtime.h>
#include <math.h>

#define BATCH 16
#define NPTS  2048
#define D4    12
#define D2    6
#define DP    16            // padded channel dim (zero rows 12..15)
#define BNEPS 1e-5f

typedef __attribute__((ext_vector_type(2))) float v2f;
typedef __attribute__((ext_vector_type(8))) float v8f;

// Inter-stage tensors live in transposed, zero-padded layout:
//   T[b][m][n] at b*DP*NPTS + m*NPTS + n,   m = 0..15, rows 12..15 == 0.
// This makes WMMA A/B fragments contiguous (one b64 load each) and makes
// per-n column reads coalesced across lanes.

// ---------------------------------------------------------------------------
// Stage 1: fc1 -> bn1 -> relu -> fc2 -> bn2 -> relu -> fc3 -> sigmoid
// One thread per n owns all 16 batches (BN normalizes per-n over (B,C)).
// ---------------------------------------------------------------------------
__global__ __launch_bounds__(256) void enc_kernel(
    const float* __restrict__ x,
    const float* __restrict__ fc1_w, const float* __restrict__ fc1_b,
    const float* __restrict__ bn1_g, const float* __restrict__ bn1_b,
    const float* __restrict__ fc2_w, const float* __restrict__ fc2_b,
    const float* __restrict__ bn2_g, const float* __restrict__ bn2_b,
    const float* __restrict__ fc3_w, const float* __restrict__ fc3_b,
    float* __restrict__ encT)
{
    const int n = blockIdx.x * 256 + threadIdx.x;

    float h1[BATCH * D2];
    #pragma unroll
    for (int b = 0; b < BATCH; ++b) {
        const float x0 = x[((size_t)b * NPTS + n) * 3 + 0];
        const float x1 = x[((size_t)b * NPTS + n) * 3 + 1];
        const float x2 = x[((size_t)b * NPTS + n) * 3 + 2];
        #pragma unroll
        for (int c = 0; c < D2; ++c) {
            float a = fc1_b[c];
            a = fmaf(x0, fc1_w[c * 3 + 0], a);
            a = fmaf(x1, fc1_w[c * 3 + 1], a);
            a = fmaf(x2, fc1_w[c * 3 + 2], a);
            h1[b * D2 + c] = a;
        }
    }
    {   // bn1 over B*D2 = 96 (training mode, biased var)
        float s = 0.f, ss = 0.f;
        #pragma unroll
        for (int i = 0; i < BATCH * D2; ++i) { s += h1[i]; ss = fmaf(h1[i], h1[i], ss); }
        const float mean = s * (1.0f / (BATCH * D2));
        const float var  = ss * (1.0f / (BATCH * D2)) - mean * mean;
        const float sc = bn1_g[n] * rsqrtf(var + BNEPS);
        const float sh = fmaf(-mean, sc, bn1_b[n]);
        #pragma unroll
        for (int i = 0; i < BATCH * D2; ++i) h1[i] = fmaxf(fmaf(h1[i], sc, sh), 0.0f);
    }
    float h2[BATCH * D4];
    #pragma unroll
    for (int b = 0; b < BATCH; ++b) {
        #pragma unroll
        for (int q = 0; q < D4; ++q) {
            float a = fc2_b[q];
            #pragma unroll
            for (int c = 0; c < D2; ++c) a = fmaf(h1[b * D2 + c], fc2_w[q * D2 + c], a);
            h2[b * D4 + q] = a;
        }
    }
    {   // bn2 over B*D4 = 192
        float s = 0.f, ss = 0.f;
        #pragma unroll
        for (int i = 0; i < BATCH * D4; ++i) { s += h2[i]; ss = fmaf(h2[i], h2[i], ss); }
        const float mean = s * (1.0f / (BATCH * D4));
        const float var  = ss * (1.0f / (BATCH * D4)) - mean * mean;
        const float sc = bn2_g[n] * rsqrtf(var + BNEPS);
        const float sh = fmaf(-mean, sc, bn2_b[n]);
        #pragma unroll
        for (int i = 0; i < BATCH * D4; ++i) h2[i] = fmaxf(fmaf(h2[i], sc, sh), 0.0f);
    }
    // fc3 + sigmoid -> transposed padded store
    #pragma unroll
    for (int b = 0; b < BATCH; ++b) {
        float* __restrict__ eT = encT + (size_t)b * DP * NPTS + n;
        #pragma unroll
        for (int d = 0; d < D4; ++d) {
            float a = fc3_b[d];
            #pragma unroll
            for (int q = 0; q < D4; ++q) a = fmaf(h2[b * D4 + q], fc3_w[d * D4 + q], a);
            eT[(size_t)d * NPTS] = 1.0f / (1.0f + expf(-a));
        }
        #pragma unroll
        for (int d = D4; d < DP; ++d) eT[(size_t)d * NPTS] = 0.0f;   // zero pad rows
    }
}

// ---------------------------------------------------------------------------
// relmod:  y = wr/N * ( s*(x_i @ (X^T U)) - s*||x_i||^2 * U_i ) + x_i
// One workgroup (8 waves) per batch; M = X^T U accumulated with
// V_WMMA_F32_16X16X4_F32 from contiguous b64 fragments, wave partials
// reduced with ds_add_f32 LDS atomics.
// ---------------------------------------------------------------------------
__global__ __launch_bounds__(256) void relmod_kernel(
    const float* __restrict__ XTall, float* __restrict__ YTall,
    float* __restrict__ UTall,
    const float* __restrict__ uw, const float* __restrict__ ub,
    const float* __restrict__ ps, const float* __restrict__ ph,
    const float* __restrict__ wr)
{
    __shared__ float ldsM[256];  // 16x16 M tile; rows/cols >= 12 exactly zero

    const int b = blockIdx.x;
    const int t = threadIdx.x;
    const float* __restrict__ XT = XTall + (size_t)b * DP * NPTS;
    float* __restrict__ YT = YTall + (size_t)b * DP * NPTS;
    float* __restrict__ UT = UTall + (size_t)b * DP * NPTS;

    ldsM[t] = 0.0f;

    // --- Phase 1: U = relu(X @ uw^T + ub) -> transposed padded store ---
    for (int k = 0; k < 8; ++k) {
        const int n = t + 256 * k;
        float xr[D4];
        #pragma unroll
        for (int p = 0; p < D4; ++p) xr[p] = XT[(size_t)p * NPTS + n];
        #pragma unroll
        for (int m = 0; m < D4; ++m) {
            float a = ub[m];
            #pragma unroll
            for (int p = 0; p < D4; ++p) a = fmaf(xr[p], uw[m * D4 + p], a);
            UT[(size_t)m * NPTS + n] = fmaxf(a, 0.0f);
        }
        #pragma unroll
        for (int m = D4; m < DP; ++m) UT[(size_t)m * NPTS + n] = 0.0f;
    }
    __threadfence();
    __syncthreads();

    // --- Phase 2: per-wave partial M via WMMA f32 16x16x4 over n-chunks ---
    // A fragment (16x4 f32): lane (m = lane&15, h = lane>>4) holds
    // K = 2h, 2h+1 -> XT[m][n0+2h], XT[m][n0+2h+1]: one aligned b64 load.
    const int wave = t >> 5;
    const int lane = t & 31;
    const int m    = lane & 15;
    const int h    = lane >> 4;

    const float* __restrict__ pA = XT + (size_t)m * NPTS + 2 * h;
    const float* __restrict__ pB = UT + (size_t)m * NPTS + 2 * h;

    v8f acc = {0.f, 0.f, 0.f, 0.f, 0.f, 0.f, 0.f, 0.f};
    const int nbase = wave * 256;            // 256 n-rows per wave
    for (int it = 0; it < 16; ++it) {        // 16 x (4 WMMAs of K=4)
        const int o = nbase + it * 16;
        v2f a0 = *(const v2f*)(pA + o);      v2f b0 = *(const v2f*)(pB + o);
        v2f a1 = *(const v2f*)(pA + o + 4);  v2f b1 = *(const v2f*)(pB + o + 4);
        v2f a2 = *(const v2f*)(pA + o + 8);  v2f b2 = *(const v2f*)(pB + o + 8);
        v2f a3 = *(const v2f*)(pA + o + 12); v2f b3 = *(const v2f*)(pB + o + 12);
        acc = __builtin_amdgcn_wmma_f32_16x16x4_f32(false, a0, false, b0, (short)0, acc, false, false);
        acc = __builtin_amdgcn_wmma_f32_16x16x4_f32(false, a1, false, b1, (short)0, acc, false, false);
        acc = __builtin_amdgcn_wmma_f32_16x16x4_f32(false, a2, false, b2, (short)0, acc, false, false);
        acc = __builtin_amdgcn_wmma_f32_16x16x4_f32(false, a3, false, b3, (short)0, acc, false, false);
    }
    // reduce 8 wave partials: D element v of lane -> row v+8h, col m
    #pragma unroll
    for (int v = 0; v < 8; ++v) {
        atomicAdd(&ldsM[(v + 8 * h) * 16 + m], acc[v]);
    }
    __syncthreads();

    // --- Phase 3: y = K*(x@M - ||x||^2 * U) + x -> transposed padded store ---
    const float Kf = wr[0] * ps[0] * ph[0] * (1.0f / (float)NPTS);
    for (int k = 0; k < 8; ++k) {
        const int n = t + 256 * k;
        float xr[D4], ur[D4];
        float xx = 0.0f;
        #pragma unroll
        for (int p = 0; p < D4; ++p) {
            xr[p] = XT[(size_t)p * NPTS + n];
            xx = fmaf(xr[p], xr[p], xx);
        }
        #pragma unroll
        for (int p = 0; p < D4; ++p) ur[p] = UT[(size_t)p * NPTS + n];
        #pragma unroll
        for (int d = 0; d < D4; ++d) {
            float s = 0.0f;
            #pragma unroll
            for (int p = 0; p < D4; ++p) s = fmaf(xr[p], ldsM[p * 16 + d], s);
            s -= xx * ur[d];
            YT[(size_t)d * NPTS + n] = fmaf(Kf, s, xr[d]);
        }
        #pragma unroll
        for (int d = D4; d < DP; ++d) YT[(size_t)d * NPTS + n] = 0.0f;
    }
}

// ---------------------------------------------------------------------------
// Stage 3: fc4 -> bn4 -> relu -> fc5 -> relu -> fc6|fc7 concat
// ---------------------------------------------------------------------------
__global__ __launch_bounds__(256) void head_kernel(
    const float* __restrict__ RT,
    const float* __restrict__ fc4_w, const float* __restrict__ fc4_b,
    const float* __restrict__ bn4_g, const float* __restrict__ bn4_b,
    const float* __restrict__ fc5_w, const float* __restrict__ fc5_b,
    const float* __restrict__ fc6_w, const float* __restrict__ fc6_b,
    const float* __restrict__ fc7_w, const float* __restrict__ fc7_b,
    float* __restrict__ out)
{
    const int n = blockIdx.x * 256 + threadIdx.x;

    float h4[BATCH * D2];
    #pragma unroll
    for (int b = 0; b < BATCH; ++b) {
        float r[D4];
        #pragma unroll
        for (int p = 0; p < D4; ++p)
            r[p] = RT[(size_t)b * DP * NPTS + (size_t)p * NPTS + n];
        #pragma unroll
        for (int c = 0; c < D2; ++c) {
            float a = fc4_b[c];
            #pragma unroll
            for (int p = 0; p < D4; ++p) a = fmaf(r[p], fc4_w[c * D4 + p], a);
            h4[b * D2 + c] = a;
        }
    }
    {   // bn4 over 96
        float s = 0.f, ss = 0.f;
        #pragma unroll
        for (int i = 0; i < BATCH * D2; ++i) { s += h4[i]; ss = fmaf(h4[i], h4[i], ss); }
        const float mean = s * (1.0f / (BATCH * D2));
        const float var  = ss * (1.0f / (BATCH * D2)) - mean * mean;
        const float sc = bn4_g[n] * rsqrtf(var + BNEPS);
        const float sh = fmaf(-mean, sc, bn4_b[n]);
        #pragma unroll
        for (int i = 0; i < BATCH * D2; ++i) h4[i] = fmaxf(fmaf(h4[i], sc, sh), 0.0f);
    }
    #pragma unroll
    for (int b = 0; b < BATCH; ++b) {
        float h5[3];
        #pragma unroll
        for (int q = 0; q < 3; ++q) {
            float a = fc5_b[q];
            #pragma unroll
            for (int c = 0; c < D2; ++c) a = fmaf(h4[b * D2 + c], fc5_w[q * D2 + c], a);
            h5[q] = fmaxf(a, 0.0f);
        }
        float o0 = fc6_b[0], o1 = fc7_b[0], o2 = fc7_b[1];
        #pragma unroll
        for (int k = 0; k < 3; ++k) {
            o0 = fmaf(h5[k], fc6_w[k], o0);
            o1 = fmaf(h5[k], fc7_w[k], o1);
            o2 = fmaf(h5[k], fc7_w[3 + k], o2);
        }
        const size_t o = ((size_t)b * NPTS + n) * 3;
        out[o + 0] = o0;
        out[o + 1] = o1;
        out[o + 2] = o2;
    }
}

// ---------------------------------------------------------------------------
extern "C" void kernel_launch(void* const* d_in, const int* in_sizes, int n_in,
                              void* d_out, int out_size, void* d_ws, size_t ws_size,
                              hipStream_t stream)
{
    const float* x     = (const float*)d_in[0];
    const float* fc1_w = (const float*)d_in[1];  const float* fc1_b = (const float*)d_in[2];
    const float* bn1_g = (const float*)d_in[3];  const float* bn1_b = (const float*)d_in[4];
    const float* fc2_w = (const float*)d_in[5];  const float* fc2_b = (const float*)d_in[6];
    const float* bn2_g = (const float*)d_in[7];  const float* bn2_b = (const float*)d_in[8];
    const float* fc3_w = (const float*)d_in[9];  const float* fc3_b = (const float*)d_in[10];
    const float* u_w[4] = {(const float*)d_in[11], (const float*)d_in[16],
                           (const float*)d_in[21], (const float*)d_in[26]};
    const float* u_b[4] = {(const float*)d_in[12], (const float*)d_in[17],
                           (const float*)d_in[22], (const float*)d_in[27]};
    const float* psv[4] = {(const float*)d_in[13], (const float*)d_in[18],
                           (const float*)d_in[23], (const float*)d_in[28]};
    const float* phv[4] = {(const float*)d_in[14], (const float*)d_in[19],
                           (const float*)d_in[24], (const float*)d_in[29]};
    const float* wrv[4] = {(const float*)d_in[15], (const float*)d_in[20],
                           (const float*)d_in[25], (const float*)d_in[30]};
    const float* fc4_w = (const float*)d_in[31]; const float* fc4_b = (const float*)d_in[32];
    const float* bn4_g = (const float*)d_in[33]; const float* bn4_b = (const float*)d_in[34];
    const float* fc5_w = (const float*)d_in[35]; const float* fc5_b = (const float*)d_in[36];
    const float* fc6_w = (const float*)d_in[37]; const float* fc6_b = (const float*)d_in[38];
    const float* fc7_w = (const float*)d_in[39]; const float* fc7_b = (const float*)d_in[40];

    const size_t TENSOR = (size_t)BATCH * DP * NPTS;    // 524288 elements (2 MB)
    float* bufT0 = (float*)d_ws;
    float* bufT1 = bufT0 + TENSOR;
    float* UbufT = bufT1 + TENSOR;                      // 6 MB of ws total

    dim3 blk(256);
    enc_kernel<<<dim3(NPTS / 256), blk, 0, stream>>>(
        x, fc1_w, fc1_b, bn1_g, bn1_b, fc2_w, fc2_b, bn2_g, bn2_b,
        fc3_w, fc3_b, bufT0);

    float* src = bufT0; float* dst = bufT1;
    for (int i = 0; i < 4; ++i) {
        relmod_kernel<<<dim3(BATCH), blk, 0, stream>>>(
            src, dst, UbufT, u_w[i], u_b[i], psv[i], phv[i], wrv[i]);
        float* tmp = src; src = dst; dst = tmp;
    }
    head_kernel<<<dim3(NPTS / 256), blk, 0, stream>>>(
        src, fc4_w, fc4_b, bn4_g, bn4_b, fc5_w, fc5_b,
        fc6_w, fc6_b, fc7_w, fc7_b, (float*)d_out);
}